// LambdaRankModel_18262200943133
// MI455X (gfx1250) — compile-verified
//
#include <hip/hip_runtime.h>
#include <hip/hip_bf16.h>

typedef float v2f __attribute__((ext_vector_type(2)));
typedef float v8f __attribute__((ext_vector_type(8)));
typedef unsigned int u32x4 __attribute__((ext_vector_type(4)));
typedef int i32x4 __attribute__((ext_vector_type(4)));
typedef int i32x8 __attribute__((ext_vector_type(8)));

#define Bsz 256
#define Nsz 256
#define Dsz 136
#define Hsz 128
#define ROWS (Bsz * Nsz)
#define K1 (Dsz / 4)      // 34 k-steps for GEMM1
#define K2 (Hsz / 4)      // 32 k-steps for GEMM2
#define NT (Hsz / 16)     // 8 column tiles
#define W1ELEMS (K1 * NT * 64)   // 17408 floats of packed W1 fragments
#define W2ELEMS (K2 * NT * 64)   // 16384 floats of packed W2 fragments
#define HBUF_FLOATS (16 * Hsz)   // 2048 floats (8KB) per-wave h buffer
#define WAVES_PER_BLOCK 8
#define SMEM_FLOATS (W1ELEMS + WAVES_PER_BLOCK * HBUF_FLOATS)  // 33792
#define SMEM_BYTES (SMEM_FLOATS * 4)                           // 135168

#if defined(__has_builtin)
#if __has_builtin(__builtin_amdgcn_tensor_load_to_lds)
#define HAVE_TDM 1
#endif
#endif

// ---------------------------------------------------------------------------
// Repack a (K x 128) row-major weight matrix into WMMA B-fragment order:
// fragment (kk,t) = 4x16 tile; lane l<16 holds {W[4kk+0][t*16+l], W[4kk+1][..]},
// lane l>=16 holds {W[4kk+2][t*16+(l-16)], W[4kk+3][..]}. float2 per lane,
// contiguous, so one 256B ds_load_b64/global_load_b64 fetches a fragment.
// ---------------------------------------------------------------------------
__global__ void pack_b_kernel(const float* __restrict__ W,
                              float* __restrict__ dst, int ksteps) {
  int idx = blockIdx.x * blockDim.x + threadIdx.x;
  int total = ksteps * NT * 32;
  if (idx >= total) return;
  int l  = idx & 31;
  int t  = (idx >> 5) & 7;
  int kk = idx >> 8;                       // NT*32 == 256
  int k   = 4 * kk + ((l & 16) ? 2 : 0);
  int col = t * 16 + (l & 15);
  dst[2 * idx]     = W[(size_t)k * Hsz + col];
  dst[2 * idx + 1] = W[(size_t)(k + 1) * Hsz + col];
}

#ifdef HAVE_TDM
// 1-D TDM copy: nelem f32 elements from global `src` to LDS byte offset
// `lds_off`. Builds a D# (group0 + group1; groups 2/3 zero / unused since
// tile_dim1==0) and issues TENSOR_LOAD_TO_LDS. Completion via TENSORcnt.
__device__ __forceinline__ void tdm_copy_1d(const float* src, unsigned lds_off,
                                            int nelem) {
  unsigned long long ga = (unsigned long long)(uintptr_t)src;
  u32x4 g0;
  g0.x = 1u;                                   // count=1 (valid), user mode
  g0.y = lds_off;                              // LDS byte address
  g0.z = (unsigned)(ga & 0xFFFFFFFFu);         // global_addr[31:0]
  g0.w = (unsigned)((ga >> 32) & 0x01FFFFFFu)  // global_addr[56:32]
         | (2u << 30);                         // type=2 ("image")
  i32x8 g1;
  g1[0] = (int)(2u << 16);                               // data_size=4B
  g1[1] = (int)(((unsigned)nelem & 0xFFFFu) << 16);      // tensor_dim0[15:0]
  g1[2] = (int)(((unsigned)nelem >> 16) | (1u << 16));   // dim0[31:16]|dim1=1
  g1[3] = (int)(((unsigned)nelem & 0xFFFFu) << 16);      // tile_dim0
  g1[4] = 0;                                             // tile_dim1/2 unused
  g1[5] = nelem;                                         // dim0_stride[31:0]
  g1[6] = 0;
  g1[7] = 0;
  i32x4 z4 = {0, 0, 0, 0};
  i32x8 z8 = {0, 0, 0, 0, 0, 0, 0, 0};
  // 6-arg form (clang-23 / therock-10.0 headers): groups 2,3 + extra group.
  __builtin_amdgcn_tensor_load_to_lds(g0, g1, z4, z4, z8, 0);
}
#endif

// ---------------------------------------------------------------------------
// Fused 3-layer MLP. 8 waves/block; wave w owns rows (block*8+w)*16..+15.
// Packed weights staged once per block into a shared LDS region by the TDM
// (W1, then region reused for W2). h1/h2 bounce through a per-wave 8KB
// XOR-swizzled LDS buffer (conflict-free stores and A-fragment loads).
// ---------------------------------------------------------------------------
__global__ void __launch_bounds__(256)
scores_kernel(const float* __restrict__ feat,
              const float* __restrict__ b1,
              const float* __restrict__ b2,
              const float* __restrict__ W3,
              const float* __restrict__ b3,
              const float* __restrict__ w1p,
              const float* __restrict__ w2p,
              float* __restrict__ scores) {
  extern __shared__ float smem[];
  float* lds_w = smem;                               // 17408 floats (69.6KB)
  const int wave = threadIdx.x >> 5;
  const int lane = threadIdx.x & 31;
  const int half = (lane >> 4) & 1;
  const int lrow = lane & 15;
  float* hbuf = smem + W1ELEMS + wave * HBUF_FLOATS; // 8KB per wave
  const int r0 = (blockIdx.x * WAVES_PER_BLOCK + wave) * 16;

  // ---- stage packed W1 into LDS (TDM DMA issued by wave 0) ----------------
#ifdef HAVE_TDM
  if (wave == 0) {
    tdm_copy_1d(w1p, (unsigned)(uintptr_t)lds_w, W1ELEMS);
    __builtin_amdgcn_s_wait_tensorcnt(0);
  }
#else
  for (int i = threadIdx.x; i < W1ELEMS; i += blockDim.x) lds_w[i] = w1p[i];
#endif
  __syncthreads();

  // ------------------ GEMM1: h1 = relu(feat(16x136) @ W1 + b1) -------------
  v8f acc[NT] = {};
  for (int kk = 0; kk < K1; ++kk) {
    const int k = 4 * kk + 2 * half;
    float2 af = *(const float2*)(feat + (size_t)(r0 + lrow) * Dsz + k);
    v2f a; a.x = af.x; a.y = af.y;
#pragma unroll
    for (int t = 0; t < NT; ++t) {
      float2 bf = *(const float2*)(lds_w + ((kk * NT + t) * 32 + lane) * 2);
      v2f bb; bb.x = bf.x; bb.y = bf.y;
      acc[t] = __builtin_amdgcn_wmma_f32_16x16x4_f32(
          false, a, false, bb, (short)0, acc[t], false, false);
    }
  }
  // bias + relu, write h1 to per-wave swizzled buffer
#pragma unroll
  for (int t = 0; t < NT; ++t) {
    float bias = b1[t * 16 + lrow];
#pragma unroll
    for (int v = 0; v < 8; ++v) {
      float x = acc[t][v] + bias;
      x = x > 0.0f ? x : 0.0f;
      const int row = v + 8 * half;
      const int col = t * 16 + lrow;
      hbuf[row * Hsz + (col ^ ((row & 15) << 2))] = x;
    }
  }
  __syncthreads();  // all GEMM1 reads of lds_w done -> safe to overwrite

  // ---- stage packed W2 into LDS (reuse region) ----------------------------
#ifdef HAVE_TDM
  if (wave == 0) {
    tdm_copy_1d(w2p, (unsigned)(uintptr_t)lds_w, W2ELEMS);
    __builtin_amdgcn_s_wait_tensorcnt(0);
  }
#else
  for (int i = threadIdx.x; i < W2ELEMS; i += blockDim.x) lds_w[i] = w2p[i];
#endif
  __syncthreads();

  // ------------------ GEMM2: h2 = relu(h1(16x128) @ W2 + b2) ---------------
  v8f acc2[NT] = {};
  for (int kk = 0; kk < K2; ++kk) {
    const int k = 4 * kk + 2 * half;
    float2 af = *(const float2*)(hbuf + lrow * Hsz + (k ^ (lrow << 2)));
    v2f a; a.x = af.x; a.y = af.y;
#pragma unroll
    for (int t = 0; t < NT; ++t) {
      float2 bf = *(const float2*)(lds_w + ((kk * NT + t) * 32 + lane) * 2);
      v2f bb; bb.x = bf.x; bb.y = bf.y;
      acc2[t] = __builtin_amdgcn_wmma_f32_16x16x4_f32(
          false, a, false, bb, (short)0, acc2[t], false, false);
    }
  }
  __syncthreads();
  // bias + relu, write h2 (reuse per-wave buffer)
#pragma unroll
  for (int t = 0; t < NT; ++t) {
    float bias = b2[t * 16 + lrow];
#pragma unroll
    for (int v = 0; v < 8; ++v) {
      float x = acc2[t][v] + bias;
      x = x > 0.0f ? x : 0.0f;
      const int row = v + 8 * half;
      const int col = t * 16 + lrow;
      hbuf[row * Hsz + (col ^ ((row & 15) << 2))] = x;
    }
  }
  __syncthreads();

  // ------------------ GEMM3: scores = h2 @ W3 + b3 -------------------------
  if (lane < 16) {
    float s = b3[0];
#pragma unroll 8
    for (int c = 0; c < Hsz; ++c)
      s += hbuf[lane * Hsz + (c ^ (lane << 2))] * W3[c];
    scores[r0 + lane] = s;
  }
}

// ---------------------------------------------------------------------------
// Pairwise lambda kernel: one block per batch row, thread i accumulates
// sum_j A[i][j] - A[j][i] with A_ji = (0.5 - lam_ij) * dndcg (shared sigmoid).
// ---------------------------------------------------------------------------
__global__ void __launch_bounds__(256)
lambda_kernel(const float* __restrict__ scores,
              const int* __restrict__ labels,
              float* __restrict__ out) {
  __shared__ float s_sh[Nsz];
  __shared__ float g_sh[Nsz];
  __shared__ int   l_sh[Nsz];
  const int b = blockIdx.x;
  const int i = threadIdx.x;

  const float si = scores[(size_t)b * Nsz + i];
  const int   li = labels[(size_t)b * Nsz + i];
  const float gi = exp2f((float)li);
  s_sh[i] = si;
  l_sh[i] = li;
  g_sh[i] = gi;
  __syncthreads();

  float acc = 0.0f;
  for (int j = 0; j < Nsz; ++j) {
    const float sj = s_sh[j];
    const int   lj = l_sh[j];
    if (li == lj) continue;                 // gain diff == 0 -> A=0 both ways
    const float mx  = fmaxf(si, sj);
    const float d   = fabsf(gi - g_sh[j]) / (1.0f + log2f(1.0f + mx));
    const float lam = 0.5f / (1.0f + expf(sj - si)); // 0.5*sigmoid(si-sj)
    acc += (li > lj) ? (lam * d) : (-(0.5f - lam) * d);
  }
  out[(size_t)b * Nsz + i] = acc;
}

// ---------------------------------------------------------------------------
extern "C" void kernel_launch(void* const* d_in, const int* in_sizes, int n_in,
                              void* d_out, int out_size, void* d_ws, size_t ws_size,
                              hipStream_t stream) {
  (void)in_sizes; (void)n_in; (void)out_size; (void)ws_size;
  const float* feat   = (const float*)d_in[0];
  const int*   labels = (const int*)d_in[1];
  const float* W1     = (const float*)d_in[2];
  const float* b1     = (const float*)d_in[3];
  const float* W2     = (const float*)d_in[4];
  const float* b2     = (const float*)d_in[5];
  const float* W3     = (const float*)d_in[6];
  const float* b3     = (const float*)d_in[7];

  float* ws     = (float*)d_ws;
  float* w1p    = ws;                    // 17408 floats
  float* w2p    = w1p + W1ELEMS;         // 16384 floats
  float* scores = w2p + W2ELEMS;         // 65536 floats

  pack_b_kernel<<<(K1 * NT * 32 + 255) / 256, 256, 0, stream>>>(W1, w1p, K1);
  pack_b_kernel<<<(K2 * NT * 32 + 255) / 256, 256, 0, stream>>>(W2, w2p, K2);
  scores_kernel<<<ROWS / (16 * WAVES_PER_BLOCK), 256, SMEM_BYTES, stream>>>(
      feat, b1, b2, W3, b3, w1p, w2p, scores);
  lambda_kernel<<<Bsz, 256, 0, stream>>>(scores, labels, (float*)d_out);
}